// SequentialSSM_65429531788051
// MI455X (gfx1250) — compile-verified
//
#include <hip/hip_runtime.h>

// Problem constants (fixed by the reference setup).
#define B_   64
#define DI   384
#define LL   197
#define NS   16
#define TPB  128
#define DTILES (DI / TPB)   // 3 d-tiles of 128

typedef unsigned int       u32;
typedef unsigned long long u64;
typedef unsigned int uint32x4 __attribute__((ext_vector_type(4)));
typedef int          int32x4  __attribute__((ext_vector_type(4)));
typedef int          int32x8  __attribute__((ext_vector_type(8)));

__device__ __forceinline__ float4 state_update(float4 a, float4 h, float4 b, float xv) {
    float4 r;
    r.x = fmaf(a.x, h.x, b.x * xv);
    r.y = fmaf(a.y, h.y, b.y * xv);
    r.z = fmaf(a.z, h.z, b.z * xv);
    r.w = fmaf(a.w, h.w, b.w * xv);
    return r;
}

__device__ __forceinline__ float dot4(float4 h, float4 c) {
    return fmaf(h.x, c.x, fmaf(h.y, c.y, fmaf(h.z, c.z, h.w * c.w)));
}

__global__ __launch_bounds__(TPB) void ssm_scan_kernel(
    const float* __restrict__ x,     // (B, DI, LL)
    const float* __restrict__ dA,    // (B, DI, LL, NS)
    const float* __restrict__ dB,    // (B, DI, LL, NS)
    const float* __restrict__ C,     // (B, NS, LL)
    const float* __restrict__ Dvec,  // (DI,)
    float* __restrict__ out)         // (B, DI, LL)
{
    __shared__ float c_raw[NS * LL];   // TDM destination: C[b] row-major [n][l]
    __shared__ float c_t[LL * NS];     // transposed [l][n] for b128 broadcast reads

    const int b  = blockIdx.x / DTILES;
    const int dt = blockIdx.x % DTILES;
    const int d  = dt * TPB + threadIdx.x;

    // ---- Tensor Data Mover: async DMA of C[b] (16 x 197 f32) into LDS ----
    // Issued by the first wave only (TDM is a wave-level op, EXEC ignored).
    if (threadIdx.x < 32) {
        const u64 gaddr = (u64)(uintptr_t)(C + (size_t)b * NS * LL);
        const u32 lds   = (u32)(uintptr_t)(&c_raw[0]);   // low 32 bits = LDS byte offset

        uint32x4 g0;
        g0[0] = 1u;                                       // count=1, is_restore=0, gather off
        g0[1] = lds;                                      // lds_addr [63:32]
        g0[2] = (u32)(gaddr & 0xFFFFFFFFull);             // global_addr [95:64]
        g0[3] = (u32)((gaddr >> 32) & 0x1FFFFFFull)       // global_addr [120:96]
              | (2u << 30);                               // type=2 ("image")

        int32x8 g1;
        g1[0] = (int)(2u << 16);                          // data_size=2 (4 bytes), mask=0
        g1[1] = (int)((u32)LL << 16);                     // tensor_dim0[15:0]=197 in [31:16]
        g1[2] = (int)((u32)NS << 16);                     // dim0 hi=0 | tensor_dim1[15:0]=16
        g1[3] = (int)((u32)LL << 16);                     // dim1 hi=0 | tile_dim0=197
        g1[4] = (int)(u32)NS;                             // tile_dim1=16, tile_dim2=0
        g1[5] = (int)(u32)LL;                             // tensor_dim0_stride=197 (elems)
        g1[6] = 0;                                        // stride hi / dim1_stride lo
        g1[7] = 0;

        int32x4 gz4 = {0, 0, 0, 0};                       // groups 2/3 unused (2-D tensor)
        int32x8 gz8 = {0, 0, 0, 0, 0, 0, 0, 0};           // extra group (clang-23 6-arg form)
        __builtin_amdgcn_tensor_load_to_lds(g0, g1, gz4, gz4, gz8, /*cpol=*/0);
        __builtin_amdgcn_s_wait_tensorcnt(0);             // TENSORcnt -> 0: DMA landed
    }
    __syncthreads();                                      // publish LDS to all waves

    // ---- transpose C[b] in LDS: [n][l] -> [l][n] (one-time, ~3K elems) ----
    for (int idx = threadIdx.x; idx < NS * LL; idx += TPB) {
        const int n = idx / LL;
        const int l = idx - n * LL;
        c_t[l * NS + n] = c_raw[idx];
    }
    __syncthreads();

    // ---- per-thread sequential scan over L, h[16] in registers ----
    const size_t bd = (size_t)b * DI + d;
    const float* __restrict__ pA = dA + bd * (size_t)(LL * NS);
    const float* __restrict__ pB = dB + bd * (size_t)(LL * NS);
    const float* __restrict__ pX = x   + bd * LL;
    float*       __restrict__ pO = out + bd * LL;
    const float  Dd = Dvec[d];

    float4 h0 = {0.f, 0.f, 0.f, 0.f};
    float4 h1 = h0, h2 = h0, h3 = h0;
    const float4* __restrict__ ct4 = (const float4*)c_t;

#pragma unroll 2
    for (int l = 0; l < LL; ++l) {
        const float4* a4 = (const float4*)(pA + (size_t)l * NS);  // 64B contiguous
        const float4* b4 = (const float4*)(pB + (size_t)l * NS);
        float4 a0 = a4[0], a1 = a4[1], a2 = a4[2], a3 = a4[3];
        float4 q0 = b4[0], q1 = b4[1], q2 = b4[2], q3 = b4[3];
        const float xv = pX[l];

        // stream prefetch ~768B/lane ahead (gfx1250 global_prefetch_b8)
        if (l + 12 < LL) {
            __builtin_prefetch(pA + (size_t)(l + 12) * NS, 0, 0);
            __builtin_prefetch(pB + (size_t)(l + 12) * NS, 0, 0);
        }

        // h_t = deltaA .* h_{t-1} + deltaB .* x_t
        h0 = state_update(a0, h0, q0, xv);
        h1 = state_update(a1, h1, q1, xv);
        h2 = state_update(a2, h2, q2, xv);
        h3 = state_update(a3, h3, q3, xv);

        // y_t = h_t . C_t  (C from LDS, uniform address -> broadcast b128 reads)
        const float4 c0 = ct4[l * 4 + 0];
        const float4 c1 = ct4[l * 4 + 1];
        const float4 c2 = ct4[l * 4 + 2];
        const float4 c3 = ct4[l * 4 + 3];
        const float y = (dot4(h0, c0) + dot4(h1, c1)) + (dot4(h2, c2) + dot4(h3, c3));

        pO[l] = fmaf(xv, Dd, y);   // + x * D skip connection
    }
}

extern "C" void kernel_launch(void* const* d_in, const int* in_sizes, int n_in,
                              void* d_out, int out_size, void* d_ws, size_t ws_size,
                              hipStream_t stream) {
    const float* x    = (const float*)d_in[0];
    const float* dA   = (const float*)d_in[1];
    const float* dB   = (const float*)d_in[2];
    const float* C    = (const float*)d_in[3];
    const float* Dvec = (const float*)d_in[4];
    float* out = (float*)d_out;

    dim3 grid(B_ * DTILES);   // 192 blocks x 128 threads = 24576 chains
    ssm_scan_kernel<<<grid, TPB, 0, stream>>>(x, dA, dB, C, Dvec, out);
}